// BalanceNLLLoss_72164040507522
// MI455X (gfx1250) — compile-verified
//
#include <hip/hip_runtime.h>
#include <stdint.h>

typedef unsigned int u32;
typedef float v2f __attribute__((ext_vector_type(2)));
typedef float v8f __attribute__((ext_vector_type(8)));

#define NB   512                     // blocks per streaming pass
#define TPB  256                     // 8 waves (wave32)
#define PIXELS (64 * 512 * 512)      // 16,777,216
#define P4     (PIXELS / 4)          // 4,194,304 float4 pixels
#define HW4    (512 * 512 / 4)       // 65,536 float4 per plane

// ---------------------------------------------------------------- math
__device__ __forceinline__ void pixnll(float x0, float x1, float& nll0, float& nll1) {
    // nll0 = softplus(x1-x0), nll1 = softplus(x0-x1), both >= 0
    float d = x1 - x0;
    float e = __expf(-fabsf(d));
    float l = __logf(1.0f + e);
    nll0 = fmaxf(d, 0.0f) + l;
    nll1 = fmaxf(-d, 0.0f) + l;
}

// Wave32 sum via V_WMMA_F32_16X16X4_F32: A[m][k0]=v(lane m), A[m][k2]=v(lane m+16),
// B = ones -> D[m][n] = v[m]+v[m+16]. D vgpr j (lanes 0-15) = v[j]+v[j+16],
// (lanes 16-31) = v[j+8]+v[j+24]. Sum 8 vgprs + xor-16 shuffle = full wave sum.
__device__ __forceinline__ float waveReduceWMMA(float v) {
    v2f a; a[0] = v;    a[1] = 0.0f;
    v2f b; b[0] = 1.0f; b[1] = 1.0f;
    v8f c = {};
    c = __builtin_amdgcn_wmma_f32_16x16x4_f32(false, a, false, b, (short)0, c, false, false);
    float s = ((c[0] + c[1]) + (c[2] + c[3])) + ((c[4] + c[5]) + (c[6] + c[7]));
    s += __shfl_xor(s, 16, 32);
    return s;
}

// ---------------------------------------------------------------- pass 1
// Full compute: loss_pos, neg_sum, N, level-1 histogram of bits[31:21] of log_neg.
__global__ void k_pass1(const float* __restrict__ in, const int* __restrict__ tgt,
                        u32* __restrict__ histC, float* __restrict__ scf, u32* __restrict__ scu) {
    __shared__ u32 h[4 * 2048];      // 4-way replicated to cut LDS atomic conflicts
    const int tid = threadIdx.x;
    for (int i = tid; i < 4 * 2048; i += TPB) h[i] = 0u;
    __syncthreads();
    u32* myh = &h[((tid >> 5) & 3) * 2048];

    const float4* in4 = (const float4*)in;
    const int4*   t4  = (const int4*)tgt;
    float accPos = 0.f, accNeg = 0.f, cntPos = 0.f;
    u32 zc = 0u;                     // exact-zero pool entries (all positives)

#define PROC1(X0, X1, T)                                                  \
    {                                                                     \
        float n0, n1; pixnll((X0), (X1), n0, n1);                         \
        float ln;                                                         \
        if ((T) != 0) { accPos += n1; cntPos += 1.0f; ln = 0.0f; }        \
        else          { accNeg += n0; ln = n0; }                          \
        u32 bits = __float_as_uint(ln);                                   \
        if (bits == 0u) zc++;                                             \
        else atomicAdd(&myh[bits >> 21], 1u);                             \
    }

    for (int q = blockIdx.x * TPB + tid; q < P4; q += NB * TPB) {
        int bb = q >> 16;
        int r  = q & 65535;
        float4 a0 = in4[(bb << 17) + r];
        float4 a1 = in4[(bb << 17) + HW4 + r];
        int4   t  = t4[q];
        PROC1(a0.x, a1.x, t.x)
        PROC1(a0.y, a1.y, t.y)
        PROC1(a0.z, a1.z, t.z)
        PROC1(a0.w, a1.w, t.w)
    }
#undef PROC1
    atomicAdd(&myh[0], zc);

    float wPos = waveReduceWMMA(accPos);
    float wNeg = waveReduceWMMA(accNeg);
    float wCnt = waveReduceWMMA(cntPos);
    if ((tid & 31) == 0) {
        atomicAdd(&scf[0], wPos);                 // loss_pos
        atomicAdd(&scf[1], wNeg);                 // sum nll0 over negatives (for CE)
        atomicAdd(&scu[2], (u32)(wCnt + 0.5f));   // N
    }
    __syncthreads();
    u32* g = &histC[(size_t)blockIdx.x * 2048];
    for (int i = tid; i < 2048; i += TPB)
        g[i] = h[i] + h[i + 2048] + h[i + 4096] + h[i + 6144];
}

// ---------------------------------------------------------------- pass 2
// S1 = sum of values with top11 > b1; level-2 histogram bits[20:10] within bucket b1.
__global__ void k_pass2(const float* __restrict__ in, const int* __restrict__ tgt,
                        u32* __restrict__ histC, float* __restrict__ scf,
                        const u32* __restrict__ scu) {
    __shared__ u32 h[2048];
    const int tid = threadIdx.x;
    for (int i = tid; i < 2048; i += TPB) h[i] = 0u;
    __syncthreads();
    const u32 b1 = scu[3] >> 21;

    const float4* in4 = (const float4*)in;
    const int4*   t4  = (const int4*)tgt;
    float accS = 0.f;
    u32 zc = 0u;

#define PROC2(X0, X1, T)                                                  \
    {                                                                     \
        float n0, n1; pixnll((X0), (X1), n0, n1);                         \
        float ln = ((T) != 0) ? 0.0f : n0;                                \
        u32 bits = __float_as_uint(ln);                                   \
        u32 top  = bits >> 21;                                            \
        if (top > b1) accS += ln;                                         \
        else if (top == b1) {                                             \
            if (bits == 0u) zc++;                                         \
            else atomicAdd(&h[(bits >> 10) & 2047u], 1u);                 \
        }                                                                 \
    }

    for (int q = blockIdx.x * TPB + tid; q < P4; q += NB * TPB) {
        int bb = q >> 16;
        int r  = q & 65535;
        float4 a0 = in4[(bb << 17) + r];
        float4 a1 = in4[(bb << 17) + HW4 + r];
        int4   t  = t4[q];
        PROC2(a0.x, a1.x, t.x)
        PROC2(a0.y, a1.y, t.y)
        PROC2(a0.z, a1.z, t.z)
        PROC2(a0.w, a1.w, t.w)
    }
#undef PROC2
    atomicAdd(&h[0], zc);            // zc nonzero only when b1 == 0

    float wS = waveReduceWMMA(accS);
    if ((tid & 31) == 0) atomicAdd(&scf[5], wS);
    __syncthreads();
    u32* g = &histC[(size_t)blockIdx.x * 2048];
    for (int i = tid; i < 2048; i += TPB) g[i] = h[i];
}

// ---------------------------------------------------------------- pass 3
// S2 = sum within bucket b1 above sub-bucket b2; level-3 count+sum hist of bits[9:0].
__global__ void k_pass3(const float* __restrict__ in, const int* __restrict__ tgt,
                        u32* __restrict__ histC, float* __restrict__ histS,
                        float* __restrict__ scf, const u32* __restrict__ scu) {
    __shared__ u32   hc[1024];
    __shared__ float hs[1024];
    const int tid = threadIdx.x;
    for (int i = tid; i < 1024; i += TPB) { hc[i] = 0u; hs[i] = 0.0f; }
    __syncthreads();
    const u32 pre   = scu[3];
    const u32 pre10 = pre >> 10;
    const u32 b1    = pre >> 21;
    const u32 b2    = (pre >> 10) & 2047u;

    const float4* in4 = (const float4*)in;
    const int4*   t4  = (const int4*)tgt;
    float accS = 0.f;
    u32 zc = 0u;

#define PROC3(X0, X1, T)                                                  \
    {                                                                     \
        float n0, n1; pixnll((X0), (X1), n0, n1);                         \
        float ln = ((T) != 0) ? 0.0f : n0;                                \
        u32 bits = __float_as_uint(ln);                                   \
        if ((bits >> 10) == pre10) {                                      \
            if (bits == 0u) zc++;                                         \
            else {                                                        \
                u32 k = bits & 1023u;                                     \
                atomicAdd(&hc[k], 1u);                                    \
                atomicAdd(&hs[k], ln);                                    \
            }                                                             \
        } else if ((bits >> 21) == b1 && ((bits >> 10) & 2047u) > b2) {   \
            accS += ln;                                                   \
        }                                                                 \
    }

    for (int q = blockIdx.x * TPB + tid; q < P4; q += NB * TPB) {
        int bb = q >> 16;
        int r  = q & 65535;
        float4 a0 = in4[(bb << 17) + r];
        float4 a1 = in4[(bb << 17) + HW4 + r];
        int4   t  = t4[q];
        PROC3(a0.x, a1.x, t.x)
        PROC3(a0.y, a1.y, t.y)
        PROC3(a0.z, a1.z, t.z)
        PROC3(a0.w, a1.w, t.w)
    }
#undef PROC3
    atomicAdd(&hc[0], zc);

    float wS = waveReduceWMMA(accS);
    if ((tid & 31) == 0) atomicAdd(&scf[5], wS);   // scf[5] accumulates S1+S2
    __syncthreads();
    u32*   gc = &histC[(size_t)blockIdx.x * 2048];
    float* gs = &histS[(size_t)blockIdx.x * 1024];
    for (int i = tid; i < 1024; i += TPB) { gc[i] = hc[i]; gs[i] = hs[i]; }
}

// ---------------------------------------------------- histogram column reduce
__global__ void k_colred_u32(const u32* __restrict__ src, u32 stride, u32 nblk, u32 n,
                             u32* __restrict__ dst) {
    u32 i = blockIdx.x * blockDim.x + threadIdx.x;
    if (i < n) {
        u32 s = 0u;
        for (u32 b = 0; b < nblk; b++) s += src[(size_t)b * stride + i];
        dst[i] = s;
    }
}

__global__ void k_colred_f32(const float* __restrict__ src, u32 stride, u32 nblk, u32 n,
                             float* __restrict__ dst) {
    u32 i = blockIdx.x * blockDim.x + threadIdx.x;
    if (i < n) {
        float s = 0.0f;
        for (u32 b = 0; b < nblk; b++) s += src[(size_t)b * stride + i];
        dst[i] = s;
    }
}

// ---------------------------------------------------------------- scan/select
// Single block. Suffix-scan counts (and sums at final level), locate boundary
// bucket for 'want', update prefix/remaining. Level 2 emits the final loss.
__global__ void k_scansel(const u32* __restrict__ cnt, const float* __restrict__ sums,
                          int nb, int shift, int level,
                          float* __restrict__ scf, u32* __restrict__ scu,
                          float* __restrict__ out) {
    __shared__ u32   sc_[2048];
    __shared__ float ss_[2048];
    const int tid = threadIdx.x;
    for (int i = tid; i < 2048; i += TPB) {
        sc_[i] = (i < nb) ? cnt[i] : 0u;
        ss_[i] = (level == 2 && i < nb) ? sums[i] : 0.0f;
    }
    __syncthreads();
    // Hillis-Steele inclusive suffix scan over 2048 entries
    for (int off = 1; off < 2048; off <<= 1) {
        u32 av[8]; float af[8];
        for (int k = 0; k < 8; k++) {
            int i = tid + (k << 8);
            av[k] = (i + off < 2048) ? sc_[i + off] : 0u;
            af[k] = (i + off < 2048) ? ss_[i + off] : 0.0f;
        }
        __syncthreads();
        for (int k = 0; k < 8; k++) {
            int i = tid + (k << 8);
            sc_[i] += av[k];
            ss_[i] += af[k];
        }
        __syncthreads();
    }
    const u32 want = (level == 0) ? scu[2] : scu[4];
    for (int k = 0; k < 8; k++) {
        int i = tid + (k << 8);
        if (i < nb) {
            u32 sufi = sc_[i];
            u32 sufn = (i + 1 < 2048) ? sc_[i + 1] : 0u;
            bool last = (i == nb - 1);
            if (sufi >= want && (last || sufn < want)) {      // unique boundary
                u32 above = last ? 0u : sufn;
                u32 R     = want - above;
                if (level < 2) {
                    scu[3] = (level == 0) ? ((u32)i << shift) : (scu[3] | ((u32)i << shift));
                    scu[4] = R;
                } else {
                    float S3 = last ? 0.0f : ss_[i + 1];
                    u32   T  = scu[3] | (u32)i;               // exact threshold pattern
                    float lossNeg = scf[5] + S3 + (float)R * __uint_as_float(T);
                    float lossPos = scf[0];
                    float Nf = (float)scu[2];
                    float ce = (lossPos + scf[1]) * (1.0f / (float)PIXELS);
                    out[0] = (lossPos + lossNeg) / (2.0f * Nf) + ce;
                }
            }
        }
    }
}

// ---------------------------------------------------------------- launcher
extern "C" void kernel_launch(void* const* d_in, const int* in_sizes, int n_in,
                              void* d_out, int out_size, void* d_ws, size_t ws_size,
                              hipStream_t stream) {
    const float* in  = (const float*)d_in[0];
    const int*   tgt = (const int*)d_in[1];
    float* out = (float*)d_out;

    char* base = (char*)d_ws;
    float* scf = (float*)base;     // [0] loss_pos  [1] neg_sum  [5] S_sel
    u32*   scu = (u32*)base;       // [2] N  [3] prefix  [4] remaining
    u32*   histC  = (u32*)(base + 256);
    float* histS  = (float*)(base + 256 + (size_t)NB * 2048 * 4);
    u32*   cntTot = (u32*)(base + 256 + (size_t)NB * 2048 * 4 + (size_t)NB * 1024 * 4);
    float* sumTot = (float*)(cntTot + 2048);

    hipMemsetAsync(d_ws, 0, 256, stream);

    // level 1: bits[31:21]
    k_pass1<<<NB, TPB, 0, stream>>>(in, tgt, histC, scf, scu);
    k_colred_u32<<<2048 / TPB, TPB, 0, stream>>>(histC, 2048, NB, 2048, cntTot);
    k_scansel<<<1, TPB, 0, stream>>>(cntTot, sumTot, 2048, 21, 0, scf, scu, out);

    // level 2: bits[20:10] within b1, plus S1
    k_pass2<<<NB, TPB, 0, stream>>>(in, tgt, histC, scf, scu);
    k_colred_u32<<<2048 / TPB, TPB, 0, stream>>>(histC, 2048, NB, 2048, cntTot);
    k_scansel<<<1, TPB, 0, stream>>>(cntTot, sumTot, 2048, 10, 1, scf, scu, out);

    // level 3: bits[9:0] within (b1,b2) with value sums, plus S2; final result
    k_pass3<<<NB, TPB, 0, stream>>>(in, tgt, histC, histS, scf, scu);
    k_colred_u32<<<1024 / TPB, TPB, 0, stream>>>(histC, 2048, NB, 1024, cntTot);
    k_colred_f32<<<1024 / TPB, TPB, 0, stream>>>(histS, 1024, NB, 1024, sumTot);
    k_scansel<<<1, TPB, 0, stream>>>(cntTot, sumTot, 1024, 0, 2, scf, scu, out);
}